// GBCE_90125593739502
// MI455X (gfx1250) — compile-verified
//
#include <hip/hip_runtime.h>
#include <math.h>

// ---------------------------------------------------------------------------
// Top-K(64) hardest-negative cross-entropy with label smoothing.
// B=4096 rows, V=50257 logits/row, fp32. Memory-bound: 823MB -> ~35us floor
// at 23.3 TB/s. Strategy: one 1024-thread block per row, stage the whole row
// (201KB) in CDNA5's 320KB LDS via async global->LDS DMA (ASYNCcnt path,
// global_load_async_to_lds_b128 via inline asm), then a 2-level radix-select
// (2048-bin LDS histograms + suffix scan) entirely out of LDS. No matmul in
// the dataflow -> WMMA not applicable; the CDNA5 wins here are 320KB LDS,
// the async-to-LDS DMA engine, and wave32 shuffle reductions.
// ---------------------------------------------------------------------------

namespace {
constexpr int   kB       = 4096;
constexpr int   kV       = 50257;
constexpr int   kK       = 64;
constexpr float kEps     = 0.1f;
constexpr int   kThreads = 1024;   // 32 waves (wave32)
constexpr int   kMaxC    = 4096;   // level-1 candidate cap
constexpr int   kMaxC2   = 512;    // level-2 candidate cap
}

// Monotone float -> uint key (ascending order preserved).
__device__ __forceinline__ unsigned f2k(float f) {
  unsigned u = __float_as_uint(f);
  return u ^ ((u & 0x80000000u) ? 0xFFFFFFFFu : 0x80000000u);
}
__device__ __forceinline__ float k2f(unsigned k) {
  unsigned u = (k & 0x80000000u) ? (k ^ 0x80000000u) : ~k;
  return __uint_as_float(u);
}

// In-place inclusive suffix scan of 2048 bins, 1024 threads (2 bins/thread).
__device__ __forceinline__ void suffix_scan2048(unsigned* h) {
  const int t = threadIdx.x;
  for (int off = 1; off < 2048; off <<= 1) {
    unsigned a0 = (t + off < 2048) ? h[t + off] : 0u;
    unsigned a1 = (t + 1024 + off < 2048) ? h[t + 1024 + off] : 0u;
    __syncthreads();
    h[t]        += a0;
    h[t + 1024] += a1;
    __syncthreads();
  }
}

// Block-wide sum of two floats; thread 0 accumulates into *accA/*accB.
__device__ __forceinline__ void block_add2(float a, float b, float* rA, float* rB,
                                           float* accA, float* accB) {
#pragma unroll
  for (int o = 16; o > 0; o >>= 1) { a += __shfl_down(a, o, 32); b += __shfl_down(b, o, 32); }
  const int lane = threadIdx.x & 31;
  const int w    = threadIdx.x >> 5;
  if (lane == 0) { rA[w] = a; rB[w] = b; }
  __syncthreads();
  if (threadIdx.x < 32) {
    a = rA[threadIdx.x]; b = rB[threadIdx.x];
#pragma unroll
    for (int o = 16; o > 0; o >>= 1) { a += __shfl_down(a, o, 32); b += __shfl_down(b, o, 32); }
    if (threadIdx.x == 0) { *accA += a; *accB += b; }
  }
  __syncthreads();
}

__global__ void __launch_bounds__(kThreads)
topk_ce_rows(const float* __restrict__ inp, const int* __restrict__ tgt,
             float* __restrict__ row_loss) {
  __shared__ __align__(16) float rowv[kV + 3];     // 201KB: whole row in LDS
  __shared__ unsigned hist[2048];                  // reused for both levels
  __shared__ unsigned candk[kMaxC];
  __shared__ float    candv[kMaxC];
  __shared__ unsigned c2k[kMaxC2];
  __shared__ float    c2v[kMaxC2];
  __shared__ float    redA[32], redB[32];
  __shared__ unsigned redU[32];
  __shared__ unsigned sh_maxkey, sh_above, sh_above2, sh_ncand, sh_nc2;
  __shared__ int      sh_b, sh_b2;
  __shared__ float    sh_sumv, sh_sume;

  const int row = blockIdx.x;
  const int t   = threadIdx.x;
  const float* __restrict__ g = inp + (size_t)row * (size_t)kV;
  const int tgti = tgt[row];

  // ---- stage row: async DMA global -> LDS (ASYNCcnt), 16B per lane.
  // LDS side is 16B aligned (j*4 bytes, j % 4 == 0); global side is 4B
  // aligned (row stride 50257), legal for >=DWORD ops in DWORD align mode.
  for (int j = t * 4; j < kV; j += kThreads * 4) {
    if (j + 4 <= kV) {
#ifdef __HIP_DEVICE_COMPILE__
      // Low 32 bits of a flat pointer into the LDS aperture == LDS byte
      // address (ISA 10.2), so no address-space casts are needed.
      unsigned lds_off = (unsigned)(size_t)(void*)&rowv[j];
      const float* gp  = g + j;
      asm volatile("global_load_async_to_lds_b128 %0, %1, off"
                   :: "v"(lds_off), "v"(gp)
                   : "memory");
#else
      rowv[j]     = g[j];
      rowv[j + 1] = g[j + 1];
      rowv[j + 2] = g[j + 2];
      rowv[j + 3] = g[j + 3];
#endif
    } else {
      for (int e = j; e < kV; ++e) rowv[e] = g[e];   // ragged tail (1 elem)
    }
  }
  for (int i = t; i < 2048; i += kThreads) hist[i] = 0u;
  if (t == 0) {
    sh_maxkey = 0u; sh_ncand = 0u; sh_nc2 = 0u;
    sh_b = -1; sh_b2 = -1; sh_above = 0u; sh_above2 = 0u;
    sh_sumv = 0.f; sh_sume = 0.f;
  }
#ifdef __HIP_DEVICE_COMPILE__
#if __has_builtin(__builtin_amdgcn_s_wait_asynccnt)
  __builtin_amdgcn_s_wait_asynccnt(0);
#else
  asm volatile("s_wait_asynccnt 0x0" ::: "memory");
#endif
#endif
  __syncthreads();

  // ---- pass 1: 2048-bin histogram of top-11 key bits + running max ----
  unsigned kmax = 0u;
  for (int j = t; j < kV; j += kThreads) {
    if (j == tgti) continue;                 // mask the true class
    unsigned k = f2k(rowv[j]);
    kmax = (k > kmax) ? k : kmax;
    atomicAdd(&hist[k >> 21], 1u);
  }
#pragma unroll
  for (int o = 16; o > 0; o >>= 1) {
    unsigned o2 = __shfl_down(kmax, o, 32);
    kmax = (o2 > kmax) ? o2 : kmax;
  }
  if ((t & 31) == 0) redU[t >> 5] = kmax;
  __syncthreads();
  if (t < 32) {
    kmax = redU[t];
#pragma unroll
    for (int o = 16; o > 0; o >>= 1) {
      unsigned o2 = __shfl_down(kmax, o, 32);
      kmax = (o2 > kmax) ? o2 : kmax;
    }
    if (t == 0) sh_maxkey = kmax;
  }
  __syncthreads();

  // ---- level-1 threshold bin: suffix scan, find S[b]>=K, S[b+1]<K ----
  suffix_scan2048(hist);
  for (int i = t; i < 2048; i += kThreads) {
    unsigned s  = hist[i];
    unsigned s1 = (i < 2047) ? hist[i + 1] : 0u;
    if (s >= (unsigned)kK && s1 < (unsigned)kK) { sh_b = i; sh_above = s1; }
  }
  __syncthreads();

  const float truth = rowv[tgti];
  const float m     = fmaxf(truth, k2f(sh_maxkey));
  const int   b     = sh_b;

  // ---- pass 2: accumulate definite winners, collect boundary-bin cands ----
  float sv = 0.f, se = 0.f;
  for (int j = t; j < kV; j += kThreads) {
    if (j == tgti) continue;
    float v = rowv[j];
    unsigned k = f2k(v);
    int bin = (int)(k >> 21);
    if (bin > b) {
      sv += v;
      se += expf(v - m);
    } else if (bin == b) {
      unsigned idx = atomicAdd(&sh_ncand, 1u);
      if (idx < (unsigned)kMaxC) { candk[idx] = k; candv[idx] = v; }
    }
  }
  block_add2(sv, se, redA, redB, &sh_sumv, &sh_sume);

  // ---- level-2 refine on next 11 key bits of the candidates ----
  const unsigned need = (unsigned)kK - sh_above;   // >= 1 by construction
  for (int i = t; i < 2048; i += kThreads) hist[i] = 0u;
  __syncthreads();
  unsigned nc = sh_ncand; if (nc > (unsigned)kMaxC) nc = (unsigned)kMaxC;
  for (unsigned i = t; i < nc; i += kThreads)
    atomicAdd(&hist[(candk[i] >> 10) & 0x7FFu], 1u);
  __syncthreads();
  suffix_scan2048(hist);
  for (int i = t; i < 2048; i += kThreads) {
    unsigned s  = hist[i];
    unsigned s1 = (i < 2047) ? hist[i + 1] : 0u;
    if (s >= need && s1 < need) { sh_b2 = i; sh_above2 = s1; }
  }
  __syncthreads();

  const int b2 = sh_b2;
  float sv2 = 0.f, se2 = 0.f;
  if (b2 >= 0) {
    for (unsigned i = t; i < nc; i += kThreads) {
      unsigned k = candk[i];
      int bin2 = (int)((k >> 10) & 0x7FFu);
      if (bin2 > b2) {
        float v = candv[i];
        sv2 += v;
        se2 += expf(v - m);
      } else if (bin2 == b2) {
        unsigned idx = atomicAdd(&sh_nc2, 1u);
        if (idx < (unsigned)kMaxC2) { c2k[idx] = k; c2v[idx] = candv[i]; }
      }
    }
  }
  block_add2(sv2, se2, redA, redB, &sh_sumv, &sh_sume);

  // ---- level-3: tiny serial arg-max selection (typically 1 element) ----
  if (t == 0) {
    unsigned n3 = sh_nc2; if (n3 > (unsigned)kMaxC2) n3 = (unsigned)kMaxC2;
    int need3 = (b2 >= 0) ? (int)need - (int)sh_above2 : 0;
    if (need3 < 0) need3 = 0;
    float sv3 = 0.f, se3 = 0.f;
    for (int s = 0; s < need3; ++s) {
      int bi = -1; unsigned bk = 0u;
      for (unsigned j2 = 0; j2 < n3; ++j2) {
        unsigned kk = c2k[j2];
        if (kk > bk) { bk = kk; bi = (int)j2; }
      }
      if (bi < 0) break;
      sv3 += c2v[bi];
      se3 += expf(c2v[bi] - m);
      c2k[bi] = 0u;                       // mark taken
    }
    const float sum_v = sh_sumv + sv3;                     // sum of top-64
    const float sum_e = sh_sume + se3 + expf(truth - m);   // softmax mass
    const float lse   = m + logf(sum_e);
    // per_ex = lse - (1-eps)*truth - (eps/(K+1))*(truth + sum_top64)
    row_loss[row] = lse - (1.0f - kEps) * truth
                        - (kEps / (float)(kK + 1)) * (truth + sum_v);
  }
}

// Deterministic fixed-tree mean over the 4096 per-row losses.
__global__ void __launch_bounds__(kThreads)
mean4096(const float* __restrict__ rl, float* __restrict__ out) {
  __shared__ float red[32];
  const int t = threadIdx.x;
  float v = rl[t] + rl[t + 1024] + rl[t + 2048] + rl[t + 3072];
#pragma unroll
  for (int o = 16; o > 0; o >>= 1) v += __shfl_down(v, o, 32);
  if ((t & 31) == 0) red[t >> 5] = v;
  __syncthreads();
  if (t < 32) {
    v = red[t];
#pragma unroll
    for (int o = 16; o > 0; o >>= 1) v += __shfl_down(v, o, 32);
    if (t == 0) out[0] = v / (float)kB;
  }
}

extern "C" void kernel_launch(void* const* d_in, const int* in_sizes, int n_in,
                              void* d_out, int out_size, void* d_ws, size_t ws_size,
                              hipStream_t stream) {
  (void)in_sizes; (void)n_in; (void)out_size; (void)ws_size;
  const float* inp = (const float*)d_in[0];
  const int*   tgt = (const int*)d_in[1];
  float*       out = (float*)d_out;
  float*       rl  = (float*)d_ws;   // 4096 per-row losses

  topk_ce_rows<<<kB, kThreads, 0, stream>>>(inp, tgt, rl);
  mean4096<<<1, kThreads, 0, stream>>>(rl, out);
}